// PhongRender_5506148073632
// MI455X (gfx1250) — compile-verified
//
#include <hip/hip_runtime.h>
#include <math.h>
#include <stddef.h>

// ---------------------------------------------------------------------------
// Phong soft rasterizer for MI455X (gfx1250).
// Outputs (concatenated flat in d_out):
//   [0]                imrender : (B,H,W,3)
//   [B*H*W*3]          improb   : (B,H,W,1)
//   [B*H*W*4]          normal1  : (B,F,3)
//
// Strategy:
//  * Kernel 1: per-face setup -> affine coefficients for (w0,w1,w2,z) as
//    functions of (PX,PY), i.e. rows [A,B,C,0] with K=4 basis [PX,PY,1,0].
//  * Kernel 2: rasterize with V_WMMA_F32_16X16X4_F32: one WMMA evaluates
//    4 faces x 16 pixels of barycentrics + interpolated depth in fp32.
//    Soft-coverage (segment distances, exp/log) stays in VALU.
// ---------------------------------------------------------------------------

typedef __attribute__((ext_vector_type(2))) float v2f;
typedef __attribute__((ext_vector_type(8))) float v8f;

#if defined(__has_builtin)
#  if __has_builtin(__builtin_amdgcn_wmma_f32_16x16x4_f32)
#    define HAVE_WMMA_F32_16X16X4 1
#  endif
#endif

#define FMAX 800                 // faces fit fully in LDS (800*20*4 = 64000 B)
#define RREC 20                  // raster record floats per face
#define SREC 18                  // shading record floats per face

__device__ __forceinline__ float clampf(float x, float lo, float hi) {
    return fminf(fmaxf(x, lo), hi);
}

__device__ __forceinline__ float segd2(float px, float py,
                                       float ux, float uy, float vx, float vy) {
    float ex = vx - ux, ey = vy - uy;
    float dx = px - ux, dy = py - uy;
    float t = (dx * ex + dy * ey) / (ex * ex + ey * ey + 1e-12f);
    t = clampf(t, 0.0f, 1.0f);
    float rx = dx - t * ex, ry = dy - t * ey;
    return rx * rx + ry * ry;
}

// ---------------------------------------------------------------------------
// Kernel 1: per-face projection + edge setup + normal1 output
// rasterRec layout (20 f): [A0,B0,C0, A1,B1,C1, A2,B2,C2, Az,Bz,Cz,
//                           ax,ay,bx,by,cx,cy, okFlag, frontFlag]
// shadeRec layout (18 f):  [nx,ny,nz, -p3_0.xyz, -p3_1.xyz, -p3_2.xyz,
//                           uv0.uv, uv1.uv, uv2.uv]
// ---------------------------------------------------------------------------
__global__ void face_setup_kernel(const float* __restrict__ points,
                                  const int*   __restrict__ faces,
                                  const float* __restrict__ camrot,
                                  const float* __restrict__ campos,
                                  const float* __restrict__ camproj,
                                  const float* __restrict__ uv,
                                  const int*   __restrict__ ft,
                                  float* __restrict__ normal1_out,
                                  float* __restrict__ rasterRec,
                                  float* __restrict__ shadeRec,
                                  int B, int P, int F) {
    int idx = blockIdx.x * blockDim.x + threadIdx.x;
    if (idx >= B * F) return;
    int b = idx / F, f = idx % F;

    const float* R = camrot + b * 9;
    float cp0 = campos[b * 3 + 0], cp1 = campos[b * 3 + 1], cp2 = campos[b * 3 + 2];
    float pr0 = camproj[0], pr1 = camproj[1], pr2 = camproj[2];

    float p3[3][3];   // camera-space vertices
    float p2[3][2];   // projected 2D
    #pragma unroll
    for (int k = 0; k < 3; ++k) {
        int vi = faces[f * 3 + k];
        float x = points[((size_t)b * P + vi) * 3 + 0] - cp0;
        float y = points[((size_t)b * P + vi) * 3 + 1] - cp1;
        float z = points[((size_t)b * P + vi) * 3 + 2] - cp2;
        float pc0 = R[0] * x + R[1] * y + R[2] * z;
        float pc1 = R[3] * x + R[4] * y + R[5] * z;
        float pc2 = R[6] * x + R[7] * y + R[8] * z;
        p3[k][0] = pc0; p3[k][1] = pc1; p3[k][2] = pc2;
        float zz = pc2 * pr2;
        p2[k][0] = (pc0 * pr0) / zz;
        p2[k][1] = (pc1 * pr1) / zz;
    }

    // face normal = cross(v1-v0, v2-v0)
    float e1x = p3[1][0] - p3[0][0], e1y = p3[1][1] - p3[0][1], e1z = p3[1][2] - p3[0][2];
    float e2x = p3[2][0] - p3[0][0], e2y = p3[2][1] - p3[0][1], e2z = p3[2][2] - p3[0][2];
    float nx = e1y * e2z - e1z * e2y;
    float ny = e1z * e2x - e1x * e2z;
    float nz = e1x * e2y - e1y * e2x;
    float nn = sqrtf(nx * nx + ny * ny + nz * nz) + 1e-15f;
    normal1_out[(size_t)(b * F + f) * 3 + 0] = nx / nn;
    normal1_out[(size_t)(b * F + f) * 3 + 1] = ny / nn;
    normal1_out[(size_t)(b * F + f) * 3 + 2] = nz / nn;

    float ax = p2[0][0], ay = p2[0][1];
    float bx = p2[1][0], by = p2[1][1];
    float cx = p2[2][0], cy = p2[2][1];
    float area = (bx - ax) * (cy - ay) - (by - ay) * (cx - ax);
    bool  ok   = fabsf(area) > 1e-10f;
    float inv  = ok ? (1.0f / area) : 1.0f;

    float rec[RREC];
    // w0: edge(b -> c)
    { float ex = cx - bx, ey = cy - by;
      rec[0] = -ey * inv; rec[1] = ex * inv; rec[2] = (ey * bx - ex * by) * inv; }
    // w1: edge(c -> a)
    { float ex = ax - cx, ey = ay - cy;
      rec[3] = -ey * inv; rec[4] = ex * inv; rec[5] = (ey * cx - ex * cy) * inv; }
    // w2: edge(a -> b)
    { float ex = bx - ax, ey = by - ay;
      rec[6] = -ey * inv; rec[7] = ex * inv; rec[8] = (ey * ax - ex * ay) * inv; }
    // z row = w0*z0 + w1*z1 + w2*z2 folded into affine form
    float z0 = p3[0][2], z1 = p3[1][2], z2 = p3[2][2];
    rec[9]  = rec[0] * z0 + rec[3] * z1 + rec[6] * z2;
    rec[10] = rec[1] * z0 + rec[4] * z1 + rec[7] * z2;
    rec[11] = rec[2] * z0 + rec[5] * z1 + rec[8] * z2;
    rec[12] = ax; rec[13] = ay; rec[14] = bx; rec[15] = by; rec[16] = cx; rec[17] = cy;
    rec[18] = ok ? 1.0f : 0.0f;
    rec[19] = (nz > 0.0f) ? 1.0f : 0.0f;

    float* rr = rasterRec + (size_t)(b * F + f) * RREC;
    #pragma unroll
    for (int i = 0; i < RREC; ++i) rr[i] = rec[i];

    float* sr = shadeRec + (size_t)(b * F + f) * SREC;
    sr[0] = nx; sr[1] = ny; sr[2] = nz;
    #pragma unroll
    for (int k = 0; k < 3; ++k) {
        sr[3 + 3 * k + 0] = -p3[k][0];
        sr[3 + 3 * k + 1] = -p3[k][1];
        sr[3 + 3 * k + 2] = -p3[k][2];
    }
    #pragma unroll
    for (int k = 0; k < 3; ++k) {
        int ti = ft[f * 3 + k];
        sr[12 + 2 * k + 0] = uv[((size_t)b * P + ti) * 2 + 0];
        sr[12 + 2 * k + 1] = uv[((size_t)b * P + ti) * 2 + 1];
    }
}

// ---------------------------------------------------------------------------
// Kernel 2: WMMA rasterization + shading.
// Block = 512 threads = 16 waves, one 16x16 pixel tile.
// Each wave owns one pixel row (16 pixels); per WMMA it evaluates 4 faces.
//   A (16x4): 16 quantity rows = 4 faces x {w0,w1,w2,z}; K = [coefA,coefB,coefC,0]
//   B (4x16): basis [PX_n, PY_n, 1, 0] per pixel column n
//   D (16x16): D[m][n] = quantity m at pixel n
// VGPR layouts per CDNA5 ISA 7.12.2 (wave32).
// ---------------------------------------------------------------------------
__global__ __launch_bounds__(512) void raster_shade_kernel(
        const float* __restrict__ rasterRec,
        const float* __restrict__ shadeRec,
        const float* __restrict__ texture,
        const float* __restrict__ lightdirect,
        const float* __restrict__ material,
        const float* __restrict__ shininess,
        float* __restrict__ out,
        int B, int F, int H, int W, int T) {
    __shared__ float sh[FMAX * RREC];

    const int tilesX = W >> 4, tilesY = H >> 4;
    int blk = blockIdx.x;
    int b   = blk / (tilesX * tilesY);
    int trm = blk % (tilesX * tilesY);
    int ty  = trm / tilesX, tx = trm % tilesX;
    int tid = threadIdx.x;

    // ---- stage full face table for this batch into LDS ----
    int nF20 = F * RREC;
    for (int i = tid; i < nF20; i += 512)
        sh[i] = rasterRec[(size_t)b * nF20 + i];
    int nQuads = (F + 3) >> 2;
    int padEnd = nQuads * 4 * RREC;
    if (padEnd > FMAX * RREC) padEnd = FMAX * RREC;
    for (int i = nF20 + tid; i < padEnd; i += 512) sh[i] = 0.0f;
    __syncthreads();

    int wave = tid >> 5, lane = tid & 31;
    int p = lane & 15, hi = lane >> 4;
    int gx = tx * 16 + p;
    int gy = ty * 16 + wave;

    float PX = (2.0f * ((float)gx + 0.5f) - (float)W) / (float)W;
    float PY = ((float)H - 2.0f * ((float)gy + 0.5f)) / (float)H;

    // B matrix: lanes 0-15 hold K0/K1 (= PX,PY), lanes 16-31 hold K2/K3 (= 1,0)
    v2f bmat;
    bmat.x = hi ? 1.0f : PX;
    bmat.y = hi ? 0.0f : PY;

    int r     = lane & 3;          // quantity row within a face: w0,w1,w2,z
    int fqloc = (lane & 15) >> 2;  // which face of the quad this lane's A-row is
    // select masks kept as multipliers so LDS loads stay unconditional
    float hiKill = hi ? 0.0f : 1.0f;

    float bestz = -1e10f;
    int   bestf = 0;
    float bw0 = 0.0f, bw1 = 0.0f, bw2 = 0.0f;
    int   cov = 0;
    float acc = 0.0f;
    const float KPROB = 1000000.0f / 7000.0f;  // MULTIPLIER^2 / DELTA

    #pragma unroll 2
    for (int q = 0; q < nQuads; ++q) {
        int fbase = q * 4;
        v8f dmat;
#ifdef HAVE_WMMA_F32_16X16X4
        int abase = (fbase + fqloc) * RREC + 3 * r;
        // lo half needs (A,B) = sh[abase], sh[abase+1]; hi half needs (C,0).
        // Loads are unconditional; the multiply (not a select-of-load) keeps
        // the compiler from predicating the LDS read with an EXEC save/restore.
        float kA = sh[abase + 2 * hi];
        float kB = sh[abase + 1];
        v2f amat;
        amat.x = kA;
        amat.y = kB * hiKill;
        v8f cmat = {0.f, 0.f, 0.f, 0.f, 0.f, 0.f, 0.f, 0.f};
        dmat = __builtin_amdgcn_wmma_f32_16x16x4_f32(
            /*neg_a=*/false, amat, /*neg_b=*/false, bmat,
            /*c_mod=*/(short)0, cmat, /*reuse_a=*/false, /*reuse_b=*/false);
#else
        // scalar fallback replicating the D VGPR layout
        #pragma unroll
        for (int j = 0; j < 8; ++j) {
            int m  = j + 8 * hi;
            int ff = fbase + (m >> 2);
            int cb = ff * RREC + 3 * (m & 3);
            dmat[j] = sh[cb] * PX + sh[cb + 1] * PY + sh[cb + 2];
        }
#endif
        // this lane's two faces: fbase + 2*hi + {0,1}
        #pragma unroll
        for (int s = 0; s < 2; ++s) {
            int f  = fbase + 2 * hi + s;
            float w0 = dmat[4 * s + 0];
            float w1 = dmat[4 * s + 1];
            float w2 = dmat[4 * s + 2];
            float zi = dmat[4 * s + 3];
            int fb = f * RREC;
            float okf   = sh[fb + 18];
            float front = sh[fb + 19];
            bool inb    = (f < F);
            bool inside = inb && (w0 >= 0.0f) && (w1 >= 0.0f) && (w2 >= 0.0f) && (okf > 0.5f);
            bool valid  = inside && (front > 0.5f);
            float zb    = valid ? zi : -1e10f;
            bool upd    = zb > bestz;       // strict > keeps earliest on ties (argmax)
            bestz = upd ? zb : bestz;
            bestf = upd ? f  : bestf;
            bw0   = upd ? w0 : bw0;
            bw1   = upd ? w1 : bw1;
            bw2   = upd ? w2 : bw2;
            cov  |= valid ? 1 : 0;

            float ax = sh[fb + 12], ay = sh[fb + 13];
            float bx = sh[fb + 14], by = sh[fb + 15];
            float cx = sh[fb + 16], cy = sh[fb + 17];
            float d2 = fminf(segd2(PX, PY, ax, ay, bx, by),
                       fminf(segd2(PX, PY, bx, by, cx, cy),
                             segd2(PX, PY, cx, cy, ax, ay)));
            float prob = inside ? 1.0f : expf(-d2 * KPROB);
            float lg   = logf(clampf(1.0f - prob, 1e-7f, 1.0f));
            acc += inb ? lg : 0.0f;
        }
    }

    // ---- merge lane l with lane l^16 (same pixel, other face pair) ----
    float oz  = __shfl_xor(bestz, 16, 32);
    int   ofi = __shfl_xor(bestf, 16, 32);
    float o0  = __shfl_xor(bw0, 16, 32);
    float o1  = __shfl_xor(bw1, 16, 32);
    float o2  = __shfl_xor(bw2, 16, 32);
    float oa  = __shfl_xor(acc, 16, 32);
    int   oc  = __shfl_xor(cov, 16, 32);
    bool take = (oz > bestz) || ((oz == bestz) && (ofi < bestf));
    bestz = take ? oz : bestz;
    bestf = take ? ofi : bestf;
    bw0 = take ? o0 : bw0; bw1 = take ? o1 : bw1; bw2 = take ? o2 : bw2;
    acc += oa;
    cov |= oc;

    if (lane < 16) {
        size_t HW  = (size_t)H * W;
        size_t pix = ((size_t)b * H + gy) * W + gx;

        // improb = 1 - exp(sum log(clip(1-prob)))
        out[(size_t)B * HW * 3 + pix] = 1.0f - expf(acc);

        float fe[9];
        #pragma unroll
        for (int i = 0; i < 9; ++i) fe[i] = 0.0f;
        if (cov) {
            const float* sr = shadeRec + (size_t)(b * F + bestf) * SREC;
            float wsum = bw0 + bw1 + bw2;
            fe[0] = sr[0] * wsum; fe[1] = sr[1] * wsum; fe[2] = sr[2] * wsum;
            #pragma unroll
            for (int d = 0; d < 3; ++d)
                fe[3 + d] = bw0 * sr[3 + d] + bw1 * sr[6 + d] + bw2 * sr[9 + d];
            fe[6] = bw0 * sr[12] + bw1 * sr[14] + bw2 * sr[16];
            fe[7] = bw0 * sr[13] + bw1 * sr[15] + bw2 * sr[17];
            fe[8] = wsum;
        }
        // normalize normal + eye
        float nn = sqrtf(fe[0]*fe[0] + fe[1]*fe[1] + fe[2]*fe[2]) + 1e-15f;
        float nx = fe[0]/nn, ny = fe[1]/nn, nz = fe[2]/nn;
        float en = sqrtf(fe[3]*fe[3] + fe[4]*fe[4] + fe[5]*fe[5]) + 1e-15f;
        float ex = fe[3]/en, ey = fe[4]/en, ez = fe[5]/en;
        // light
        float lx = lightdirect[b*3+0], ly = lightdirect[b*3+1], lz = lightdirect[b*3+2];
        float ln = sqrtf(lx*lx + ly*ly + lz*lz) + 1e-15f;
        lx /= ln; ly /= ln; lz /= ln;

        float cosT = clampf(nx*lx + ny*ly + nz*lz, 0.0f, 1.0f);
        float rx = 2.0f*cosT*nx - lx, ry = 2.0f*cosT*ny - ly, rz = 2.0f*cosT*nz - lz;
        float cosA = clampf(rx*ex + ry*ey + rz*ez, 1e-5f, 1.0f);
        cosA = powf(cosA, shininess[b]);

        // bilinear texture fetch
        float u = clampf(fe[6], 0.0f, 1.0f) * (float)(T - 1);
        float v = (1.0f - clampf(fe[7], 0.0f, 1.0f)) * (float)(T - 1);
        float x0f = floorf(u), y0f = floorf(v);
        int x0i = (int)x0f, y0i = (int)y0f;
        int x1i = min(x0i + 1, T - 1);
        int y1i = min(y0i + 1, T - 1);
        float wx = u - x0f, wy = v - y0f;

        #pragma unroll
        for (int c = 0; c < 3; ++c) {
            const float* tp = texture + ((size_t)(b * 3 + c) * T) * T;
            float t00 = tp[(size_t)y0i * T + x0i];
            float t01 = tp[(size_t)y0i * T + x1i];
            float t10 = tp[(size_t)y1i * T + x0i];
            float t11 = tp[(size_t)y1i * T + x1i];
            float col = t00 * (1.0f - wx) * (1.0f - wy) + t01 * wx * (1.0f - wy)
                      + t10 * (1.0f - wx) * wy          + t11 * wx * wy;
            float amb  = material[b * 9 + 0 + c];
            float dif  = material[b * 9 + 3 + c];
            float spec = material[b * 9 + 6 + c];
            float colr = (amb * col + dif * cosT * col + spec * cosA) * fe[8];
            out[pix * 3 + c] = clampf(colr, 0.0f, 1.0f);
        }
    }
}

// ---------------------------------------------------------------------------
// Host launcher
// ---------------------------------------------------------------------------
extern "C" void kernel_launch(void* const* d_in, const int* in_sizes, int n_in,
                              void* d_out, int out_size, void* d_ws, size_t ws_size,
                              hipStream_t stream) {
    const float* points   = (const float*)d_in[0];
    const int*   faces    = (const int*)  d_in[1];
    const float* camrot   = (const float*)d_in[2];
    const float* campos   = (const float*)d_in[3];
    const float* camproj  = (const float*)d_in[4];
    const float* uv       = (const float*)d_in[5];
    const int*   ft       = (const int*)  d_in[6];
    const float* texture  = (const float*)d_in[7];
    const float* lightdir = (const float*)d_in[8];
    const float* material = (const float*)d_in[9];
    const float* shin     = (const float*)d_in[10];

    int B = in_sizes[3] / 3;                 // camera_pos (B,3)
    int F = in_sizes[1] / 3;                 // faces (F,3)
    int P = in_sizes[0] / (3 * B);           // points (B,P,3)
    int T = (int)(sqrt((double)(in_sizes[7] / (3 * B))) + 0.5);  // texture (B,3,T,T)
    int HWtot = (out_size - B * F * 3) / (4 * B);
    int H = (int)(sqrt((double)HWtot) + 0.5);
    int W = H;

    float* out = (float*)d_out;
    float* rasterRec = (float*)d_ws;
    float* shadeRec  = rasterRec + (size_t)B * F * RREC;
    float* normal1   = out + (size_t)B * H * W * 4;

    int nFace = B * F;
    face_setup_kernel<<<(nFace + 255) / 256, 256, 0, stream>>>(
        points, faces, camrot, campos, camproj, uv, ft,
        normal1, rasterRec, shadeRec, B, P, F);

    int nBlocks = B * (H / 16) * (W / 16);
    raster_shade_kernel<<<nBlocks, 512, 0, stream>>>(
        rasterRec, shadeRec, texture, lightdir, material, shin,
        out, B, F, H, W, T);
}